// dissected_Conv2d_79654463471825
// MI455X (gfx1250) — compile-verified
//
#include <hip/hip_runtime.h>
#include <stdint.h>

// 3x3 conv, 64->64 ch, 112x112, pad 1, fp32 — implicit GEMM on the exact
// fp32 CDNA5 matrix path V_WMMA_F32_16X16X4_F32.
//   M = B*H*W = 25088, N = 64, K = 9 taps * 64 in-ch = 576 (tap-major,
//   channel-minor so each K=4 chunk stays inside one tap).
// Block = 128 thr (4 waves) covers 2 output rows x 16 pixels x 64 out-ch.
// Wave w owns N-tile w and both row tiles (B fragment reused across rows).
// LDS layouts put ci innermost (padded to 68 dwords) so every operand
// fragment is ONE conflict-free ds_load_b64 with immediate offsets.
// Weight slab staged via gfx1250 async global->LDS DMA (ASYNCcnt path).

#define IN_C  64
#define OUT_C 64
#define BATCH 2
#define HH    112
#define WW    112
#define CI_PAD 68  // 64 + 4 pad dwords: 8B-aligned, bank-conflict-free b64

typedef __attribute__((ext_vector_type(2))) float v2f;
typedef __attribute__((ext_vector_type(8))) float v8f;
typedef __attribute__((ext_vector_type(4))) int v4i;
typedef __attribute__((address_space(1))) v4i gv4i;  // global (AS1) 16B chunk
typedef __attribute__((address_space(3))) v4i lv4i;  // LDS    (AS3) 16B chunk

#if defined(__has_builtin)
#if __has_builtin(__builtin_amdgcn_global_load_async_to_lds_b128)
#define HAVE_ASYNC_LDS 1
#else
#define HAVE_ASYNC_LDS 0
#endif
#else
#define HAVE_ASYNC_LDS 0
#endif

// ---------------------------------------------------------------------------
// Prep: weight[o][ci][kh][kw] -> wmat[p][o][ci]  (p = kh*3+kw), 147456 B.
// Makes each tap's B slab a contiguous 16KB block with ci innermost.
// ---------------------------------------------------------------------------
__global__ __launch_bounds__(256) void weight_permute_kernel(
    const float* __restrict__ weight, float* __restrict__ wmat) {
  int t = blockIdx.x * blockDim.x + threadIdx.x;  // t = p*4096 + o*64 + ci
  if (t >= 9 * 64 * 64) return;
  int ci = t & 63;
  int o  = (t >> 6) & 63;
  int p  = t >> 12;
  wmat[t] = weight[(o * 64 + ci) * 9 + p];
}

// ---------------------------------------------------------------------------
__global__ __launch_bounds__(128) void conv3x3_wmma_kernel(
    const float* __restrict__ x, const float* __restrict__ wmat,
    const float* __restrict__ bias, float* __restrict__ out) {
  __shared__ float lds_in[4 * 18 * CI_PAD];  // [r(4)][c(18)][ci(pad 68)]
  __shared__ float lds_w[64 * CI_PAD];       // [o(64)][ci(pad 68)]

  const int tid  = threadIdx.x;
  const int lane = tid & 31;
  const int wv   = tid >> 5;   // N tile (0..3)
  const int half = lane >> 4;  // K-split for A/B, M-split for D
  const int m    = lane & 15;  // A row (pixel) / B,D column

  const int blk   = blockIdx.x;
  const int strip = blk % 7;
  const int hh    = (blk / 7) % (HH / 2);
  const int b     = blk / (7 * (HH / 2));
  const int h0    = hh * 2;        // block covers output rows h0, h0+1
  const int w0    = strip * 16;

  // Stage zero-padded halo: rows h0-1..h0+2, cols w0-1..w0+16, all 64 ch.
  for (int e = tid; e < 4 * 18 * IN_C; e += 128) {
    int c  = e % 18;
    int r  = (e / 18) & 3;
    int ci = e / 72;
    int gh = h0 - 1 + r;
    int gw = w0 - 1 + c;
    float v = 0.0f;
    if ((unsigned)gh < (unsigned)HH && (unsigned)gw < (unsigned)WW)
      v = x[((b * IN_C + ci) * HH + gh) * WW + gw];
    lds_in[(r * 18 + c) * CI_PAD + ci] = v;
  }

  v8f acc0 = {}, acc1 = {};          // two 16x16 f32 tiles (rows h0, h0+1)
  const int bcol    = wv * 16 + m;   // B column (output channel within slab)
  const int ci_lane = 2 * half;      // this lane's K sub-offset

  for (int p = 0; p < 9; ++p) {
    __syncthreads();  // protect previous tap's lds_w reads
#if HAVE_ASYNC_LDS
    {  // async DMA: global -> LDS, per-lane padded destination rows
      const float* src = wmat + p * 4096;
      for (int e = tid; e < 1024; e += 128) {  // e indexes float4 chunks
        int o = e >> 4, c4 = e & 15;
        const float* g = src + e * 4;
        float* l = &lds_w[o * CI_PAD + c4 * 4];
        __builtin_amdgcn_global_load_async_to_lds_b128(
            (gv4i*)(uintptr_t)g, (lv4i*)(uint32_t)(uintptr_t)l, 0, 0);
      }
      asm volatile("s_wait_asynccnt 0x0" ::: "memory");
    }
#else
    {  // fallback: plain 16KB coalesced copy with padded rows
      const float4* src = (const float4*)(wmat + p * 4096);
      for (int e = tid; e < 1024; e += 128) {
        int o = e >> 4, c4 = e & 15;
        *(float4*)&lds_w[o * CI_PAD + c4 * 4] = src[e];
      }
    }
#endif
    if (p < 8) __builtin_prefetch(wmat + (p + 1) * 4096 + tid * 32, 0, 1);
    __syncthreads();

    const int kh = p / 3;
    const int kw = p - kh * 3;
    // Base pointers: all 16 K-steps reachable via immediate ds offsets (+16B).
    const float* ap0 = &lds_in[((kh + 0) * 18 + m + kw) * CI_PAD + ci_lane];
    const float* ap1 = &lds_in[((kh + 1) * 18 + m + kw) * CI_PAD + ci_lane];
    const float* bp  = &lds_w[bcol * CI_PAD + ci_lane];

#pragma unroll
    for (int ci0 = 0; ci0 < 64; ci0 += 4) {
      v2f A0 = *(const v2f*)(ap0 + ci0);  // one ds_load_b64 each
      v2f A1 = *(const v2f*)(ap1 + ci0);
      v2f B  = *(const v2f*)(bp + ci0);
      acc0 = __builtin_amdgcn_wmma_f32_16x16x4_f32(
          false, A0, false, B, (short)0, acc0, false, false);
      acc1 = __builtin_amdgcn_wmma_f32_16x16x4_f32(
          false, A1, false, B, (short)0, acc1, false, false);
    }
  }

  // D layout: acc[r] -> (M = half*8 + r, N = m); each lane's 8 outputs per
  // row are contiguous in w.
  const int o = wv * 16 + m;
  const float bo = bias[o];
  float* d0 = out + (((b * OUT_C + o) * HH + h0) * WW + w0 + half * 8);
  float* d1 = d0 + WW;
#pragma unroll
  for (int r = 0; r < 8; ++r) {
    d0[r] = acc0[r] + bo;
    d1[r] = acc1[r] + bo;
  }
}

// ---------------------------------------------------------------------------
extern "C" void kernel_launch(void* const* d_in, const int* in_sizes, int n_in,
                              void* d_out, int out_size, void* d_ws,
                              size_t ws_size, hipStream_t stream) {
  const float* x      = (const float*)d_in[0];
  const float* weight = (const float*)d_in[1];
  const float* bias   = (const float*)d_in[2];
  float* out  = (float*)d_out;
  float* wmat = (float*)d_ws;  // needs 9*64*64*4 = 147456 B scratch

  weight_permute_kernel<<<(9 * 64 * 64 + 255) / 256, 256, 0, stream>>>(weight,
                                                                       wmat);
  conv3x3_wmma_kernel<<<BATCH * (HH / 2) * 7, 128, 0, stream>>>(x, wmat, bias,
                                                                out);
}